// GHMC_LossV2_12403865550912
// MI455X (gfx1250) — compile-verified
//
#include <hip/hip_runtime.h>
#include <hip/hip_bf16.h>

// GHM-C loss for MI455X (gfx1250, wave32).
// Pass 1 uses V_WMMA_F32_16X16X4_F32 as a full-fp32 row-reduction engine:
//   rowsum(16 rows x K) = A(16x4 of exp(pred)) x B(4x16 of ones), accumulated.
// Histogram via integer atomics (deterministic); final weighted sum via a
// fixed-order single-block tree reduction (deterministic).

typedef __attribute__((ext_vector_type(2))) float v2f;
typedef __attribute__((ext_vector_type(8))) float v8f;

#define NBINS 11   // BINS + 1

__global__ void ghm_zero_counts(unsigned int* counts) {
    if (threadIdx.x < NBINS) counts[threadIdx.x] = 0u;
}

// One wave handles 16 rows. 256 threads = 8 waves -> 128 rows per block.
__global__ __launch_bounds__(256)
void ghm_pass1_wmma(const float* __restrict__ pred,
                    const int* __restrict__ target,
                    int C,
                    float* __restrict__ loss,
                    int* __restrict__ bins,
                    unsigned int* __restrict__ counts) {
    __shared__ unsigned int s_cnt[NBINS];
    if (threadIdx.x < NBINS) s_cnt[threadIdx.x] = 0u;
    __syncthreads();

    const int lane = threadIdx.x & 31;
    const int wave = threadIdx.x >> 5;
    const int m    = lane & 15;   // row within 16-row tile (A-matrix M index)
    const int half = lane >> 4;   // 0: K={0,1}, 1: K={2,3}  (A-matrix layout)

    const long long row_base = ((long long)blockIdx.x * 8 + wave) * 16;
    const float* __restrict__ rowp = pred + (size_t)(row_base + m) * (size_t)C;

    v8f acc0 = {0.f,0.f,0.f,0.f,0.f,0.f,0.f,0.f};
    v8f acc1 = {0.f,0.f,0.f,0.f,0.f,0.f,0.f,0.f};
    v2f bones; bones.x = 1.0f; bones.y = 1.0f;   // B = all-ones 4x16

    int c = 0;
    const int C8 = C & ~7;
    for (; c < C8; c += 8) {
        // lane<16 loads cols {c,c+1}; lane>=16 loads {c+2,c+3} of the same row
        const float2 x0 = *reinterpret_cast<const float2*>(rowp + c + 2 * half);
        const float2 x1 = *reinterpret_cast<const float2*>(rowp + c + 4 + 2 * half);
        v2f a0; a0.x = __expf(x0.x); a0.y = __expf(x0.y);
        v2f a1; a1.x = __expf(x1.x); a1.y = __expf(x1.y);
        acc0 = __builtin_amdgcn_wmma_f32_16x16x4_f32(
            false, a0, false, bones, (short)0, acc0, false, false);
        acc1 = __builtin_amdgcn_wmma_f32_16x16x4_f32(
            false, a1, false, bones, (short)0, acc1, false, false);
    }
    if (c + 3 < C) {
        const float2 x0 = *reinterpret_cast<const float2*>(rowp + c + 2 * half);
        v2f a0; a0.x = __expf(x0.x); a0.y = __expf(x0.y);
        acc0 = __builtin_amdgcn_wmma_f32_16x16x4_f32(
            false, a0, false, bones, (short)0, acc0, false, false);
        c += 4;
    }
    // leftover columns (C % 4): scalar, owned by lanes 0..15 (one row each)
    float extra = 0.0f;
    if (lane < 16) {
        for (int cc = c; cc < C; ++cc) extra += __expf(rowp[cc]);
    }

    v8f acc;
#pragma unroll
    for (int i = 0; i < 8; ++i) acc[i] = acc0[i] + acc1[i];

    // D layout: VGPR v, lanes 0-15 -> M=v ; lanes 16-31 -> M=v+8.
    // Broadcast leftover sums so lane 0 gets rows 0..7, lane 16 rows 8..15.
    float exs[8];
    const int xb = half ? 8 : 0;
#pragma unroll
    for (int k = 0; k < 8; ++k) exs[k] = __shfl(extra, xb + k, 32);

    if (lane == 0 || lane == 16) {
        const int rbase = half ? 8 : 0;
#pragma unroll
        for (int k = 0; k < 8; ++k) {
            const long long row = row_base + rbase + k;
            const float rsum = acc[k] + exs[k];
            const int   t    = target[row];
            const float ptl  = pred[(size_t)row * (size_t)C + (size_t)t];
            const float g    = 1.0f - __expf(ptl) / rsum;           // 1 - softmax[t]
            int b = (int)floorf((g + 0.05f) * 10.0f);               // (g+ALPHA)*BINS
            b = b < 0 ? 0 : (b > 10 ? 10 : b);
            loss[row] = -ptl + __logf(rsum + 1e-8f);
            bins[row] = b;
            atomicAdd(&s_cnt[b], 1u);
        }
    }
    __syncthreads();
    if (threadIdx.x < NBINS) {
        const unsigned int v = s_cnt[threadIdx.x];
        if (v) atomicAdd(&counts[threadIdx.x], v);
    }
}

// Scalar fallback for B % 128 rows (one wave per row). Not launched for the
// reference shape (131072 % 128 == 0) but keeps the kernel shape-general.
__global__ __launch_bounds__(256)
void ghm_pass1_tail(const float* __restrict__ pred,
                    const int* __restrict__ target,
                    int C, int row0, int B,
                    float* __restrict__ loss,
                    int* __restrict__ bins,
                    unsigned int* __restrict__ counts) {
    const int wave = threadIdx.x >> 5;
    const int lane = threadIdx.x & 31;
    const int row  = row0 + blockIdx.x * 8 + wave;
    if (row >= B) return;
    const float* __restrict__ rp = pred + (size_t)row * (size_t)C;
    float s = 0.0f;
    for (int c = lane; c < C; c += 32) s += __expf(rp[c]);
#pragma unroll
    for (int off = 16; off > 0; off >>= 1) s += __shfl_down(s, off, 32);
    if (lane == 0) {
        const int   t   = target[row];
        const float ptl = rp[t];
        const float g   = 1.0f - __expf(ptl) / s;
        int b = (int)floorf((g + 0.05f) * 10.0f);
        b = b < 0 ? 0 : (b > 10 ? 10 : b);
        loss[row] = -ptl + __logf(s + 1e-8f);
        bins[row] = b;
        atomicAdd(&counts[b], 1u);
    }
}

// Deterministic finalize: out = (sum_i loss_i / counts[bin_i]) / n
__global__ __launch_bounds__(1024)
void ghm_pass2(const float* __restrict__ loss,
               const int* __restrict__ bins,
               const unsigned int* __restrict__ counts,
               int B, float* __restrict__ out) {
    __shared__ float s_inv[NBINS];
    __shared__ float s_n;
    __shared__ float s_red[1024];
    if (threadIdx.x == 0) {
        int n = 0;
        for (int b = 0; b < NBINS; ++b) {
            const unsigned int cb = counts[b];
            s_inv[b] = cb ? 1.0f / (float)cb : 0.0f;
            n += (cb != 0u);
        }
        s_n = (n > 0) ? (float)n : 1.0f;
    }
    __syncthreads();
    float local = 0.0f;
    for (int i = threadIdx.x; i < B; i += 1024)
        local += loss[i] * s_inv[bins[i]];
    s_red[threadIdx.x] = local;
    __syncthreads();
    for (int s = 512; s > 0; s >>= 1) {
        if ((int)threadIdx.x < s) s_red[threadIdx.x] += s_red[threadIdx.x + s];
        __syncthreads();
    }
    if (threadIdx.x == 0) out[0] = s_red[0] / s_n;
}

extern "C" void kernel_launch(void* const* d_in, const int* in_sizes, int n_in,
                              void* d_out, int out_size, void* d_ws, size_t ws_size,
                              hipStream_t stream) {
    const float* pred   = (const float*)d_in[0];
    const int*   target = (const int*)d_in[1];
    const int B = in_sizes[1];
    const int C = in_sizes[0] / B;
    float* out = (float*)d_out;

    unsigned char* ws = (unsigned char*)d_ws;
    unsigned int* counts = (unsigned int*)ws;                      // 11 u32
    float*        loss   = (float*)(ws + 64);                      // B floats
    int*          bins   = (int*)(ws + 64 + (size_t)B * 4);        // B ints

    ghm_zero_counts<<<1, 32, 0, stream>>>(counts);

    const int nblk = B / 128;
    if (nblk > 0)
        ghm_pass1_wmma<<<nblk, 256, 0, stream>>>(pred, target, C, loss, bins, counts);

    const int rem = B - nblk * 128;
    if (rem > 0)
        ghm_pass1_tail<<<(rem + 7) / 8, 256, 0, stream>>>(pred, target, C,
                                                          nblk * 128, B,
                                                          loss, bins, counts);

    ghm_pass2<<<1, 1024, 0, stream>>>(loss, bins, counts, B, out);
}